// GeneRegulationEmbedding_57750130262642
// MI455X (gfx1250) — compile-verified
//
#include <hip/hip_runtime.h>
#include <hip/hip_bf16.h>
#include <math.h>

// ---------------------------------------------------------------------------
// GeneRegulationEmbedding for MI455X (gfx1250, wave32)
//   B=4, G=2048, D=256, V=60000
// Bandwidth-bound (~277 MB out, ~130 MB in  ->  ~17us at 23.3 TB/s).
// The einsums s_i = e . w[:D], s_j = e . w[D:] run on the matrix unit via
// V_WMMA_F32_16X16X4_F32 (A = 16x4 tile of e, B col0 = w_lo, col1 = w_hi).
// This revision removes the per-iteration EXEC divergence in the WMMA loop:
// all lanes load w unconditionally (address is valid for every lane) and the
// B operand is masked with v_cndmask instead of saveexec branches.
// ---------------------------------------------------------------------------

#define B_ 4
#define G_ 2048
#define D_ 256

typedef __attribute__((ext_vector_type(2))) float v2f;
typedef __attribute__((ext_vector_type(4))) float f4;
typedef __attribute__((ext_vector_type(8))) float v8f;

// ---- 1) embedding gather: e[b,g,:] = emb_table[gene_ids[b,g],:] -----------
__global__ void grn_gather_kernel(const int* __restrict__ gene_ids,
                                  const float* __restrict__ emb,
                                  float* __restrict__ e_out) {
  const int bg = blockIdx.x;          // 0 .. B*G-1
  const int t  = threadIdx.x;         // 0 .. 63  (64*4 = 256 = D)
  const int gid = gene_ids[bg];
  const f4* __restrict__ src = (const f4*)(emb + (size_t)gid * D_);
  f4* __restrict__ dst = (f4*)(e_out + (size_t)bg * D_);
  dst[t] = src[t];
}

// ---- 2) s_i / s_j via WMMA f32 16x16x4 ------------------------------------
// One wave (32 lanes) per 16 embedding rows.
// A layout (16x4 f32): lanes 0-15 hold M=0..15 with {K0,K1}; lanes 16-31 {K2,K3}.
// B layout (4x16 f32): lanes hold N, lane-half/vgpr hold K (mirror of A).
// B col 0 = w_lo chunk, col 1 = w_hi chunk, other cols zero.
// C (16x16): N=0 column = s_i (lanes 0/16), N=1 column = s_j (lanes 1/17).
__global__ void grn_dots_wmma_kernel(const float* __restrict__ e,
                                     const float* __restrict__ w,
                                     float* __restrict__ si,
                                     float* __restrict__ sj) {
  const int lane  = threadIdx.x;      // 0..31
  const int m     = lane & 15;
  const int khalf = lane >> 4;        // 0 -> K{0,1}, 1 -> K{2,3}
  const int rowbase = blockIdx.x * 16;

  const float* __restrict__ arow = e + (size_t)(rowbase + m) * D_ + khalf * 2;
  // Valid address for ALL lanes (m&1 in {0,1}); lanes with m>=2 load a
  // duplicate value that is masked off below -> no EXEC divergence in loop.
  const float* __restrict__ wbase = w + (m & 1) * D_ + khalf * 2;
  const bool active = (m < 2);

  v8f c = {};
#pragma unroll 4
  for (int kb = 0; kb < D_; kb += 4) {
    v2f a;
    a.x = arow[kb];
    a.y = arow[kb + 1];
    const float wx = wbase[kb];
    const float wy = wbase[kb + 1];
    v2f bv;
    bv.x = active ? wx : 0.0f;        // v_cndmask, not saveexec
    bv.y = active ? wy : 0.0f;
    // 8 args: (neg_a, A, neg_b, B, c_mod, C, reuse_a, reuse_b)
    c = __builtin_amdgcn_wmma_f32_16x16x4_f32(
        false, a, false, bv, (short)0, c, false, false);
  }

  // C layout: vgpr r, lanes 0-15 -> (M=r, N=lane); lanes 16-31 -> (M=8+r).
  if (active) {                       // lanes 0,1,16,17 carry the results
    float* __restrict__ dst = ((m == 0) ? si : sj) + rowbase + khalf * 8;
#pragma unroll
    for (int r = 0; r < 8; ++r) dst[r] = c[r];
  }
}

// ---- 3) gate / classifier / sigmoid partial sums --------------------------
__global__ void grn_gateclass_kernel(const float* __restrict__ gl,
                                     const float* __restrict__ cl,
                                     float* __restrict__ gate_out,
                                     float* __restrict__ cprob_out,
                                     float* __restrict__ cv_ws,
                                     float* __restrict__ partials) {
  const int tid = threadIdx.x;
  const int idx = blockIdx.x * 256 + tid;   // < G*G

  const float x   = gl[idx];
  const float sig = 1.0f / (1.0f + expf(-x));
  gate_out[idx]   = (x > 0.0f) ? 1.0f : 0.0f;   // sigmoid(x) > 0.5  <=>  x > 0

  const size_t c3 = (size_t)idx * 3;
  const float l0 = cl[c3 + 0];
  const float l1 = cl[c3 + 1];
  const float l2 = cl[c3 + 2];
  int am = 0;
  float best = l0;
  if (l1 > best) { best = l1; am = 1; }
  if (l2 > best) { am = 2; }
  cprob_out[c3 + 0] = (am == 0) ? 1.0f : 0.0f;
  cprob_out[c3 + 1] = (am == 1) ? 1.0f : 0.0f;
  cprob_out[c3 + 2] = (am == 2) ? 1.0f : 0.0f;
  cv_ws[idx] = (am == 0) ? -1.0f : ((am == 2) ? 1.0f : 0.0f);

  __shared__ float red[256];
  red[tid] = sig;
  __syncthreads();
  for (int s = 128; s > 0; s >>= 1) {
    if (tid < s) red[tid] += red[tid + s];
    __syncthreads();
  }
  if (tid == 0) partials[blockIdx.x] = red[0];
}

// ---- 4) deterministic final reduction of l0_reg ---------------------------
__global__ void grn_l0_reduce_kernel(const float* __restrict__ partials,
                                     int n, float* __restrict__ out) {
  __shared__ float red[256];
  const int tid = threadIdx.x;
  float s = 0.0f;
  for (int i = tid; i < n; i += 256) s += partials[i];
  red[tid] = s;
  __syncthreads();
  for (int st = 128; st > 0; st >>= 1) {
    if (tid < st) red[tid] += red[tid + st];
    __syncthreads();
  }
  if (tid == 0) out[0] = red[0];
}

// ---- 5) big streaming kernel: interaction / sparse / regulatory -----------
// One float4 (4 contiguous j) per thread; non-temporal stores keep the 192 MB
// L2 free for the gate/cv rereads across the B dimension.
__global__ void grn_stream_kernel(const float* __restrict__ si,
                                  const f4* __restrict__ sj4,
                                  const f4* __restrict__ gate4,
                                  const f4* __restrict__ cv4,
                                  const float* __restrict__ bptr,
                                  f4* __restrict__ out_inter,
                                  f4* __restrict__ out_sparse,
                                  f4* __restrict__ out_reg) {
  const int q  = blockIdx.x * 256 + threadIdx.x;  // < B*G*G/4 = 4,194,304
  const int j4 = q & 511;                         // G/4 = 512
  const int t  = q >> 9;                          // b*G + i
  const int i  = t & (G_ - 1);

  const float s_i  = si[t];
  const float bias = bptr[0];
  const f4 s_j = sj4[j4];
  const int gi = (i << 9) + j4;                   // (i,j) tile index
  const f4 g = gate4[gi];
  const f4 c = cv4[gi];

  const f4 inter = s_j + (s_i + bias);
  const f4 sp    = inter * g;
  const f4 rg    = sp * c;

  __builtin_nontemporal_store(inter, out_inter + q);
  __builtin_nontemporal_store(sp,    out_sparse + q);
  __builtin_nontemporal_store(rg,    out_reg + q);
}

// ---------------------------------------------------------------------------
extern "C" void kernel_launch(void* const* d_in, const int* in_sizes, int n_in,
                              void* d_out, int out_size, void* d_ws, size_t ws_size,
                              hipStream_t stream) {
  const int*   gene_ids = (const int*)  d_in[0];   // (B,G) int32
  const float* emb      = (const float*)d_in[1];   // (V,D)
  const float* w        = (const float*)d_in[2];   // (2D,)
  const float* bproj    = (const float*)d_in[3];   // scalar
  const float* gl       = (const float*)d_in[4];   // (G*G,)
  const float* cl       = (const float*)d_in[5];   // (G*G,3)

  float* out = (float*)d_out;
  const size_t GG = (size_t)G_ * G_;
  float* out_e      = out;                                  // B*G*D
  float* out_inter  = out_e      + (size_t)B_ * G_ * D_;    // B*G*G
  float* out_sparse = out_inter  + (size_t)B_ * GG;         // B*G*G
  float* out_reg    = out_sparse + (size_t)B_ * GG;         // B*G*G
  float* out_gate   = out_reg    + (size_t)B_ * GG;         // G*G
  float* out_cprob  = out_gate   + GG;                      // G*G*3
  float* out_l0     = out_cprob  + 3 * GG;                  // 1

  float* ws       = (float*)d_ws;
  float* si_ws    = ws;                       // B*G
  float* sj_ws    = ws + (size_t)B_ * G_;     // B*G
  float* cv_ws    = ws + (size_t)2 * B_ * G_; // G*G
  float* partials = cv_ws + GG;               // G*G/256

  const int n_gc_blocks = (int)(GG / 256);              // 16384

  grn_gather_kernel<<<B_ * G_, 64, 0, stream>>>(gene_ids, emb, out_e);

  grn_dots_wmma_kernel<<<(B_ * G_) / 16, 32, 0, stream>>>(out_e, w, si_ws, sj_ws);

  grn_gateclass_kernel<<<n_gc_blocks, 256, 0, stream>>>(gl, cl, out_gate,
                                                        out_cprob, cv_ws, partials);

  grn_l0_reduce_kernel<<<1, 256, 0, stream>>>(partials, n_gc_blocks, out_l0);

  const int n_stream_blocks = (int)((B_ * GG / 4) / 256);  // 16384
  grn_stream_kernel<<<n_stream_blocks, 256, 0, stream>>>(
      si_ws, (const f4*)sj_ws, (const f4*)out_gate, (const f4*)cv_ws, bproj,
      (f4*)out_inter, (f4*)out_sparse, (f4*)out_reg);
}